// ContrastiveNoiseAnchor_28389733827343
// MI455X (gfx1250) — compile-verified
//
#include <hip/hip_runtime.h>
#include <math.h>

#define B_N 8192
#define D_K 256
#define INV_T 10.0f          // 1 / TEMPERATURE
#define ACT_Q 0.1f
#define LDS_STRIDE 260       // 256 + 4 pad dwords -> conflict-free A-fragment reads

typedef __attribute__((ext_vector_type(2))) float v2f;
typedef __attribute__((ext_vector_type(8))) float v8f;

// ---------------- row normalization: one wave per row ----------------
__global__ __launch_bounds__(256) void nrm_kernel(const float* __restrict__ X,
                                                  float* __restrict__ Xn) {
  int row  = (blockIdx.x * blockDim.x + threadIdx.x) >> 5;
  int lane = threadIdx.x & 31;
  if (row >= B_N) return;
  const float* x = X + (size_t)row * D_K;
  float ss = 0.f;
#pragma unroll
  for (int k = lane; k < D_K; k += 32) { float v = x[k]; ss += v * v; }
#pragma unroll
  for (int off = 16; off >= 1; off >>= 1) ss += __shfl_xor(ss, off, 32);
  float inv = rsqrtf(ss);
  float* y = Xn + (size_t)row * D_K;
#pragma unroll
  for (int k = lane; k < D_K; k += 32) y[k] = x[k] * inv;
}

// ---------------- bitonic sort of 8192 floats in LDS ----------------
__global__ __launch_bounds__(1024) void sort_kernel(const float* __restrict__ in,
                                                    float* __restrict__ out) {
  __shared__ float buf[B_N];
  int tid = threadIdx.x;
  for (int i = tid; i < B_N; i += blockDim.x) buf[i] = in[i];
  __syncthreads();
  for (int k = 2; k <= B_N; k <<= 1) {
    for (int j = k >> 1; j > 0; j >>= 1) {
      for (int i = tid; i < B_N; i += blockDim.x) {
        int l = i ^ j;
        if (l > i) {
          float a = buf[i], b = buf[l];
          bool up = ((i & k) == 0);
          if (up ? (a > b) : (a < b)) { buf[i] = b; buf[l] = a; }
        }
      }
      __syncthreads();
    }
  }
  for (int i = tid; i < B_N; i += blockDim.x) out[i] = buf[i];
}

// count ordered pairs (i,j) with 0 < |S_j - S_i| <= x   (S sorted ascending)
__device__ unsigned long long count_le(const float* __restrict__ S, float x,
                                       int tid, int nthr,
                                       unsigned long long* sh) {
  if (tid == 0) *sh = 0ull;
  __syncthreads();
  unsigned long long loc = 0;
  for (int i = tid; i < B_N; i += nthr) {
    float ti = S[i];
    int a = i + 1, b = B_N;
    while (a < b) { int m = (a + b) >> 1; if (S[m] - ti <= x) a = m + 1; else b = m; }
    int up = a - (i + 1);
    int a2 = 0, b2 = i;
    while (a2 < b2) { int m = (a2 + b2) >> 1; if (ti - S[m] <= x) b2 = m; else a2 = m + 1; }
    int dn = i - a2;
    int e = i + 1, e2 = B_N;
    while (e < e2) { int m = (e + e2) >> 1; if (S[m] - ti <= 0.0f) e = m + 1; else e2 = m; }
    int equp = e - (i + 1);
    int f = 0, f2 = i;
    while (f < f2) { int m = (f + f2) >> 1; if (ti - S[m] <= 0.0f) f2 = m; else f = m + 1; }
    int eqdn = i - f;
    loc += (unsigned long long)((up - equp) + (dn - eqdn));
  }
  atomicAdd(sh, loc);
  __syncthreads();
  unsigned long long r = *sh;
  __syncthreads();
  return r;
}

__device__ float select_kth(const float* __restrict__ S, unsigned long long rank,
                            int tid, int nthr, unsigned long long* sh) {
  unsigned lo = 0u, hi = 0x7f800000u;
  while (lo < hi) {                    // uniform: count is block-shared
    unsigned mid = lo + ((hi - lo) >> 1);
    unsigned long long c = count_le(S, __uint_as_float(mid), tid, nthr, sh);
    if (c >= rank) hi = mid; else lo = mid + 1;
  }
  return __uint_as_float(lo);
}

// sc[0]=noise_thr sc[1]=act_thr sc[2]=loss_sum sc[3]=pair count (uint bits)
__global__ __launch_bounds__(1024) void thr_kernel(const float* __restrict__ sortedA,
                                                   const float* __restrict__ sortedT,
                                                   float* __restrict__ sc) {
  __shared__ unsigned long long sh;
  int tid = threadIdx.x, nthr = blockDim.x;
  unsigned long long m = count_le(sortedT, __uint_as_float(0x7f800000u), tid, nthr, &sh);
  float mf   = (float)m;
  float pos  = ACT_Q * (mf - 1.0f);
  float flo  = floorf(pos), fhi = ceilf(pos);
  float frac = pos - flo;
  unsigned long long rlo = (unsigned long long)flo + 1ull;
  unsigned long long rhi = (unsigned long long)fhi + 1ull;
  float vlo = select_kth(sortedT, rlo, tid, nthr, &sh);
  float vhi = select_kth(sortedT, rhi, tid, nthr, &sh);
  float act = vlo * (1.0f - frac) + vhi * frac;
  if (tid == 0) {
    sc[0] = 0.5f * (sortedA[B_N / 2 - 1] + sortedA[B_N / 2]);  // q=0.5 median
    sc[1] = act;
    sc[2] = 0.0f;
    ((unsigned*)sc)[3] = 0u;
  }
}

__global__ void low_kernel(const float* __restrict__ alea,
                           const float* __restrict__ sc, int* __restrict__ low) {
  int i = blockIdx.x * blockDim.x + threadIdx.x;
  if (i < B_N) low[i] = (alea[i] < sc[0]) ? 1 : 0;
}

// ---------------- TDM: stage a 16x256 f32 row tile into LDS ----------------
// D# per CDNA5 ISA: 2-D tensor (dim0=256, dim1=8192), tile 256x16, data=4B,
// pad 4 dwords every 256 dwords -> LDS row stride 260 (bank-conflict free).
__device__ __forceinline__ void tdm_load_tile(const float* gsrc, void* lds_dst) {
#if __has_builtin(__builtin_amdgcn_tensor_load_to_lds)
  typedef unsigned __attribute__((ext_vector_type(4))) u32x4;
  typedef int      __attribute__((ext_vector_type(8))) i32x8;
  typedef int      __attribute__((ext_vector_type(4))) i32x4;
  unsigned long long ga = (unsigned long long)(size_t)gsrc;
  unsigned lds_off = (unsigned)(size_t)lds_dst;     // addr[31:0] == LDS offset
  u32x4 g0; i32x8 g1; i32x4 g2 = {0, 0, 0, 0}, g3 = {0, 0, 0, 0};
  i32x8 g4 = {0, 0, 0, 0, 0, 0, 0, 0};
  g0[0] = 1u;                                              // count=1 user D#
  g0[1] = lds_off;                                         // lds_addr
  g0[2] = (unsigned)(ga & 0xFFFFFFFFull);                  // global_addr lo
  g0[3] = (unsigned)((ga >> 32) & 0x01FFFFFFull) | (2u << 30);  // hi + type=2
  g1[0] = (int)((2u << 16)       // data_size = 4 bytes
              | (1u << 20)       // pad_enable
              | (7u << 22)       // pad_interval code 7 = 256 dwords
              | (3u << 25));     // pad_amount  code 3 = 4 dwords
  g1[1] = (int)(((unsigned)D_K & 0xFFFFu) << 16);               // tensor_dim0 lo
  g1[2] = (int)(((unsigned)D_K >> 16) | (((unsigned)B_N & 0xFFFFu) << 16)); // dim0 hi|dim1 lo
  g1[3] = (int)(((unsigned)B_N >> 16) | ((unsigned)D_K << 16)); // dim1 hi|tile_dim0
  g1[4] = 16;                                                   // tile_dim1=16
  g1[5] = D_K;                                                  // tensor_dim0_stride
  g1[6] = 0; g1[7] = 0;
  // 6-arg form (clang-23 / therock headers): extra int32x8 group before cpol.
  __builtin_amdgcn_tensor_load_to_lds(g0, g1, g2, g3, g4, 0);
  __builtin_amdgcn_s_wait_tensorcnt(0);
#endif
}

// ---------------- fused epilogues (fully branchless: bitwise masks) ----------------
__device__ __forceinline__ void p1_tile(const v8f& c, int jc, float tj, int lj,
                                        const float* tRow, const int* lowRow,
                                        const int* rIdx, float act_thr,
                                        float* sS, int* pA) {
#pragma unroll
  for (int r = 0; r < 8; ++r) {
    float s  = c[r] * INV_T;
    float ad = fabsf(tRow[r] - tj);
    int   sa = (int)(ad < act_thr);
    int   negm = sa & (lowRow[r] ^ lj);                       // exactly one low
    int   posm = sa & lowRow[r] & lj & (int)(rIdx[r] != jc);  // both low, off-diag
    sS[r] += (float)negm * __expf(s);   // s in [-10,10]: no max-shift needed
    pA[r] |= posm;
  }
}

__device__ __forceinline__ void p2_tile(const v8f& c, int jc, float tj, int lj,
                                        const float* tRow, const int* lowRow,
                                        const int* rIdx, const float* nlRow,
                                        const int* rvRow, float act_thr,
                                        float& lsum, unsigned& lcnt) {
#pragma unroll
  for (int r = 0; r < 8; ++r) {
    float s  = c[r] * INV_T;
    float ad = fabsf(tRow[r] - tj);
    int take = (int)(ad < act_thr) & lowRow[r] & lj &
               (int)(rIdx[r] != jc) & rvRow[r];
    float nl  = nlRow[r];
    float lae = fmaxf(s, nl) + log1pf(__expf(-fabsf(s - nl)));  // logaddexp
    lsum += (float)take * (lae - s);
    lcnt += (unsigned)take;
  }
}

// ---------------- pass 1: per-row sum(exp) over hard negatives ----------------
// Block: 16 rows, 8 waves; each wave computes two 16x16 sim tiles per iteration
// (one A fragment feeds two WMMAs). C layout: lane(l,half), VGPR r ->
// sim(rowBase + r + 8*half, colBase + l).
__global__ __launch_bounds__(256) void pass1_kernel(
    const float* __restrict__ Xn, const float* __restrict__ tgt,
    const int* __restrict__ low, const float* __restrict__ sc,
    float* __restrict__ neg_lse, int* __restrict__ row_valid) {
  __shared__ float ldsA[16 * LDS_STRIDE];
  __shared__ float redS[8 * 16];
  __shared__ int   redP[8 * 16];
  const int tid = threadIdx.x;
  const int wv = tid >> 5, lane = tid & 31;
  const int l = lane & 15, half = lane >> 4;
  const int rowBase = blockIdx.x * 16;
  const float act_thr = sc[1];

#if __has_builtin(__builtin_amdgcn_tensor_load_to_lds)
  if (tid < 32) tdm_load_tile(Xn + (size_t)rowBase * D_K, (void*)ldsA);
#else
  for (int i = tid; i < 16 * D_K; i += blockDim.x) {
    int r = i >> 8, kk = i & 255;
    ldsA[r * LDS_STRIDE + kk] = Xn[(size_t)(rowBase + r) * D_K + kk];
  }
#endif
  __syncthreads();

  float tRow[8]; int lowRow[8]; int rIdx[8];
#pragma unroll
  for (int r = 0; r < 8; ++r) {
    int row = rowBase + r + 8 * half;
    rIdx[r] = row; tRow[r] = tgt[row]; lowRow[r] = low[row];
  }
  float sS[8]; int pA[8];
#pragma unroll
  for (int r = 0; r < 8; ++r) { sS[r] = 0.f; pA[r] = 0; }

  const float* Arow = &ldsA[l * LDS_STRIDE + 2 * half];
  const int NW = blockDim.x >> 5;

  for (int ct = wv * 2; ct < B_N / 16; ct += NW * 2) {
    int jc0 = ct * 16 + l, jc1 = jc0 + 16;
    float tj0 = tgt[jc0], tj1 = tgt[jc1];
    int lj0 = low[jc0], lj1 = low[jc1];
    const float* Bp0 = Xn + (size_t)jc0 * D_K + 2 * half;
    const float* Bp1 = Xn + (size_t)jc1 * D_K + 2 * half;
    v8f c0 = {0.f, 0.f, 0.f, 0.f, 0.f, 0.f, 0.f, 0.f};
    v8f c1 = {0.f, 0.f, 0.f, 0.f, 0.f, 0.f, 0.f, 0.f};
#pragma unroll 4
    for (int kk = 0; kk < D_K; kk += 4) {
      v2f a  = *(const v2f*)(Arow + kk);
      v2f b0 = *(const v2f*)(Bp0 + kk);
      v2f b1 = *(const v2f*)(Bp1 + kk);
      c0 = __builtin_amdgcn_wmma_f32_16x16x4_f32(false, a, false, b0, (short)0,
                                                 c0, false, false);
      c1 = __builtin_amdgcn_wmma_f32_16x16x4_f32(false, a, false, b1, (short)0,
                                                 c1, false, false);
    }
    p1_tile(c0, jc0, tj0, lj0, tRow, lowRow, rIdx, act_thr, sS, pA);
    p1_tile(c1, jc1, tj1, lj1, tRow, lowRow, rIdx, act_thr, sS, pA);
  }
  // sum / OR across the 16 lanes sharing each row (xor 1,2,4,8)
#pragma unroll
  for (int off = 1; off <= 8; off <<= 1) {
#pragma unroll
    for (int r = 0; r < 8; ++r) {
      sS[r] += __shfl_xor(sS[r], off, 32);
      pA[r] |= __shfl_xor(pA[r], off, 32);
    }
  }
  if (l == 0) {
#pragma unroll
    for (int r = 0; r < 8; ++r) {
      int slot = wv * 16 + 8 * half + r;
      redS[slot] = sS[r]; redP[slot] = pA[r];
    }
  }
  __syncthreads();
  if (tid < 16) {
    float sum = 0.f; int pa = 0;
    for (int w = 0; w < NW; ++w) { sum += redS[w * 16 + tid]; pa |= redP[w * 16 + tid]; }
    int hasneg = (sum > 0.f) ? 1 : 0;        // exp(s) > 0 always => sum>0 iff any neg
    neg_lse[rowBase + tid] = hasneg ? __logf(sum) : 0.0f;
    row_valid[rowBase + tid] = hasneg & (pa != 0 ? 1 : 0);
  }
}

// ---------------- pass 2: accumulate positive-pair loss ----------------
__global__ __launch_bounds__(256) void pass2_kernel(
    const float* __restrict__ Xn, const float* __restrict__ tgt,
    const int* __restrict__ low, const float* __restrict__ sc,
    const float* __restrict__ neg_lse, const int* __restrict__ row_valid,
    float* __restrict__ acc /* acc[0]=loss_sum, ((uint*)acc)[1]=count */) {
  __shared__ float ldsA[16 * LDS_STRIDE];
  __shared__ float redL[8];
  __shared__ unsigned redC[8];
  const int tid = threadIdx.x;
  const int wv = tid >> 5, lane = tid & 31;
  const int l = lane & 15, half = lane >> 4;
  const int rowBase = blockIdx.x * 16;
  const float act_thr = sc[1];

#if __has_builtin(__builtin_amdgcn_tensor_load_to_lds)
  if (tid < 32) tdm_load_tile(Xn + (size_t)rowBase * D_K, (void*)ldsA);
#else
  for (int i = tid; i < 16 * D_K; i += blockDim.x) {
    int r = i >> 8, kk = i & 255;
    ldsA[r * LDS_STRIDE + kk] = Xn[(size_t)(rowBase + r) * D_K + kk];
  }
#endif
  __syncthreads();

  float tRow[8], nlRow[8]; int lowRow[8], rvRow[8], rIdx[8];
#pragma unroll
  for (int r = 0; r < 8; ++r) {
    int row = rowBase + r + 8 * half;
    rIdx[r] = row; tRow[r] = tgt[row]; lowRow[r] = low[row];
    nlRow[r] = neg_lse[row]; rvRow[r] = row_valid[row];
  }
  float lsum = 0.f; unsigned lcnt = 0;

  const float* Arow = &ldsA[l * LDS_STRIDE + 2 * half];
  const int NW = blockDim.x >> 5;

  for (int ct = wv * 2; ct < B_N / 16; ct += NW * 2) {
    int jc0 = ct * 16 + l, jc1 = jc0 + 16;
    float tj0 = tgt[jc0], tj1 = tgt[jc1];
    int lj0 = low[jc0], lj1 = low[jc1];
    const float* Bp0 = Xn + (size_t)jc0 * D_K + 2 * half;
    const float* Bp1 = Xn + (size_t)jc1 * D_K + 2 * half;
    v8f c0 = {0.f, 0.f, 0.f, 0.f, 0.f, 0.f, 0.f, 0.f};
    v8f c1 = {0.f, 0.f, 0.f, 0.f, 0.f, 0.f, 0.f, 0.f};
#pragma unroll 4
    for (int kk = 0; kk < D_K; kk += 4) {
      v2f a  = *(const v2f*)(Arow + kk);
      v2f b0 = *(const v2f*)(Bp0 + kk);
      v2f b1 = *(const v2f*)(Bp1 + kk);
      c0 = __builtin_amdgcn_wmma_f32_16x16x4_f32(false, a, false, b0, (short)0,
                                                 c0, false, false);
      c1 = __builtin_amdgcn_wmma_f32_16x16x4_f32(false, a, false, b1, (short)0,
                                                 c1, false, false);
    }
    p2_tile(c0, jc0, tj0, lj0, tRow, lowRow, rIdx, nlRow, rvRow, act_thr, lsum, lcnt);
    p2_tile(c1, jc1, tj1, lj1, tRow, lowRow, rIdx, nlRow, rvRow, act_thr, lsum, lcnt);
  }
#pragma unroll
  for (int off = 1; off <= 16; off <<= 1) {
    lsum += __shfl_xor(lsum, off, 32);
    lcnt += (unsigned)__shfl_xor((int)lcnt, off, 32);
  }
  if (lane == 0) { redL[wv] = lsum; redC[wv] = lcnt; }
  __syncthreads();
  if (tid == 0) {
    float t = 0.f; unsigned cc = 0;
    for (int w = 0; w < NW; ++w) { t += redL[w]; cc += redC[w]; }
    atomicAdd(&acc[0], t);
    atomicAdd(((unsigned*)acc) + 1, cc);
  }
}

__global__ void fin_kernel(const float* __restrict__ sc, float* __restrict__ out) {
  unsigned n = ((const unsigned*)sc)[3];
  float denom = (float)(n > 0u ? n : 1u);
  out[0] = sc[2] / denom;
}

extern "C" void kernel_launch(void* const* d_in, const int* in_sizes, int n_in,
                              void* d_out, int out_size, void* d_ws, size_t ws_size,
                              hipStream_t stream) {
  const float* emb  = (const float*)d_in[0];
  const float* tgt  = (const float*)d_in[1];
  const float* alea = (const float*)d_in[2];

  char* ws = (char*)d_ws;
  float* Xn      = (float*)ws;                                  // 8 MiB
  float* sortedA = (float*)(ws + (size_t)B_N * D_K * sizeof(float));
  float* sortedT = sortedA + B_N;
  float* nlse    = sortedT + B_N;
  int*   rvalid  = (int*)(nlse + B_N);
  int*   lowm    = rvalid + B_N;
  float* sc      = (float*)(lowm + B_N);

  nrm_kernel <<<B_N / 8, 256, 0, stream>>>(emb, Xn);
  sort_kernel<<<1, 1024, 0, stream>>>(alea, sortedA);
  sort_kernel<<<1, 1024, 0, stream>>>(tgt, sortedT);
  thr_kernel <<<1, 1024, 0, stream>>>(sortedA, sortedT, sc);
  low_kernel <<<B_N / 256, 256, 0, stream>>>(alea, sc, lowm);
  pass1_kernel<<<B_N / 16, 256, 0, stream>>>(Xn, tgt, lowm, sc, nlse, rvalid);
  pass2_kernel<<<B_N / 16, 256, 0, stream>>>(Xn, tgt, lowm, sc, nlse, rvalid, sc + 2);
  fin_kernel <<<1, 1, 0, stream>>>(sc, (float*)d_out);
}